// ChamferDistanceL1_14293651161195
// MI455X (gfx1250) — compile-verified
//
#include <hip/hip_runtime.h>

typedef __attribute__((ext_vector_type(2))) float v2f;
typedef __attribute__((ext_vector_type(8))) float v8f;

#define B_  4
#define N_  8192
#define M_  8192
#define TILE 16
#define WAVES_PER_WG 8
#define ROWS_PER_WG (TILE * WAVES_PER_WG)   // 128 rows per workgroup
#define WGS_PER_BATCH (N_ / ROWS_PER_WG)    // 64
#define NT (M_ / TILE)                      // 512 column tiles (power of two)
#define INF_BITS 0x7F800000u

// Raw hardware sqrt (single v_sqrt_f32, ~1 ulp) — only used outside hot loops.
__device__ __forceinline__ float fast_sqrtf(float x) {
#if defined(__has_builtin) && __has_builtin(__builtin_amdgcn_sqrtf)
    return __builtin_amdgcn_sqrtf(x);
#else
    return __builtin_sqrtf(x);
#endif
}

__device__ __forceinline__ unsigned uminu(unsigned a, unsigned b) { return a < b ? a : b; }

// ---------------------------------------------------------------------------
// Kernel 1: initialize scalar output and global column-min scratch (+inf bits)
// ---------------------------------------------------------------------------
__global__ void chamfer_init(unsigned* __restrict__ colmin, float* __restrict__ out) {
    int i = blockIdx.x * blockDim.x + threadIdx.x;
    if (i == 0) out[0] = 0.0f;
    if (i < B_ * M_) colmin[i] = INF_BITS;
}

// ---------------------------------------------------------------------------
// Kernel 2: main WMMA kernel.
// Each wave owns a 16-row tile of xyz1 and sweeps all M columns in 16-wide
// tiles; inner products via v_wmma_f32_16x16x4_f32 (K=3 zero-padded to 4).
// Mins are taken over SQUARED distances (sqrt deferred: both sqrt and
// max(.,0) are monotonic), accumulated as unsigned bit patterns (valid for
// non-negative floats). ||xyz2||^2 is precomputed per-WG into an LDS table so
// the inner loop's LDS fetch is index-only and prefetchable a full iteration
// ahead, in parallel with the pipelined global B-tile load.
// ---------------------------------------------------------------------------
__global__ __launch_bounds__(256) void chamfer_main(const float* __restrict__ xyz1,
                                                    const float* __restrict__ xyz2,
                                                    unsigned* __restrict__ colmin,
                                                    float* __restrict__ out) {
    __shared__ unsigned smin[M_];   // 32 KB: per-WG column min-d2 (bit-ordered fp32)
    __shared__ float    ssq2[M_];   // 32 KB: ||xyz2[c]||^2 table

    const int tid  = threadIdx.x;
    const int lane = tid & 31;
    const int wave = tid >> 5;
    const int wg   = blockIdx.x;
    const int b    = wg / WGS_PER_BATCH;
    const int rowbase = (wg % WGS_PER_BATCH) * ROWS_PER_WG + wave * TILE;

    const float* __restrict__ p1 = xyz1 + (size_t)b * N_ * 3;
    const float* __restrict__ p2 = xyz2 + (size_t)b * M_ * 3;

    // Cooperative setup: +inf column mins and the ||q||^2 table.
    for (int i = tid; i < M_; i += 256) {
        smin[i] = INF_BITS;
        const float x = p2[i * 3 + 0], y = p2[i * 3 + 1], z = p2[i * 3 + 2];
        ssq2[i] = x * x + y * y + z * z;
    }
    __syncthreads();

    const int hi  = lane >> 4;   // half-wave select: K=0,1 vs K=2,3
    const int l16 = lane & 15;

    // A tile (16x4 f32 WMMA layout): lanes 0-15 -> K=0,1 ; lanes 16-31 -> K=2, pad0
    v2f A;
    {
        const int r = rowbase + l16;
        if (hi == 0) { A.x = p1[r * 3 + 0]; A.y = p1[r * 3 + 1]; }
        else         { A.x = p1[r * 3 + 2]; A.y = 0.0f;          }
    }

    // Per-lane squared norms of the 8 rows this lane's C-VGPRs cover,
    // and running row min-d2 accumulators (unsigned fp32 bit patterns).
    float    s1r[8];
    unsigned rowaccu[8];
#pragma unroll
    for (int v = 0; v < 8; ++v) {
        const int r = rowbase + v + hi * 8;
        const float x = p1[r * 3 + 0], y = p1[r * 3 + 1], z = p1[r * 3 + 2];
        s1r[v]     = x * x + y * y + z * z;
        rowaccu[v] = INF_BITS;
    }

    // Software-pipelined, branch-free fetch for tile 0:
    // B components at p2[c*3 + hi] (hi=0 sees [x,y], hi=1 sees [y,z]),
    // plus the column's ||q||^2 from the LDS table.
    float t0   = p2[l16 * 3 + hi];
    float t1   = p2[l16 * 3 + hi + 1];
    float sq2c = ssq2[l16];

    for (int ct = 0; ct < NT; ++ct) {
        // Prefetch next tile; branch-free wrap (NT is a power of two).
        const int cn = (((ct + 1) & (NT - 1)) * TILE) + l16;
        const float n0   = p2[cn * 3 + hi];
        const float n1   = p2[cn * 3 + hi + 1];
        const float sq2n = ssq2[cn];

        // B tile (4x16 f32 WMMA layout): hi=0 -> (x,y) rows K=0,1 ; hi=1 -> (z,0)
        v2f Bm;
        Bm.x = hi ? t1 : t0;
        Bm.y = hi ? 0.0f : t1;

        v8f Cz = {};
        v8f D = __builtin_amdgcn_wmma_f32_16x16x4_f32(
            /*neg_a=*/false, A, /*neg_b=*/false, Bm,
            /*c_mod=*/(short)0, Cz, /*reuse_a=*/false, /*reuse_b=*/false);

        unsigned cminu = INF_BITS;
#pragma unroll
        for (int v = 0; v < 8; ++v) {
            const float d2 = fmaf(-2.0f, D[v], s1r[v] + sq2c);
            const unsigned u = __float_as_uint(fmaxf(d2, 0.0f));  // >=0, canonical
            rowaccu[v] = uminu(rowaccu[v], u);
            cminu      = uminu(cminu, u);
        }
        // Both half-lanes hit the same column slot; ds_min_u32 merges them.
        atomicMin(&smin[ct * TILE + l16], cminu);

        t0 = n0; t1 = n1; sq2c = sq2n;
    }

    // Finalize row mins: reduce across the 16 lanes of each half-wave,
    // then take the deferred sqrt (16 sqrts per wave total).
    float rsum = 0.0f;
#pragma unroll
    for (int v = 0; v < 8; ++v) {
        unsigned m = rowaccu[v];
#pragma unroll
        for (int off = 1; off < 16; off <<= 1)
            m = uminu(m, (unsigned)__shfl_xor((int)m, off, 32));
        rsum += fast_sqrtf(__uint_as_float(m));   // uniform within each half
    }
    rsum += __shfl_xor(rsum, 16, 32);   // combine the two half-wave row groups
    if (lane == 0)
        atomicAdd(out, rsum * (1.0f / ((float)B_ * (float)N_)));

    __syncthreads();
    // Merge this WG's 128-row column min-d2 into the global per-batch colmin.
    for (int i = tid; i < M_; i += 256)
        atomicMin(&colmin[(size_t)b * M_ + i], smin[i]);
}

// ---------------------------------------------------------------------------
// Kernel 3: sqrt + sum the B*M column minima into the scalar output.
// ---------------------------------------------------------------------------
__global__ void chamfer_cols(const unsigned* __restrict__ colmin, float* __restrict__ out) {
    const int i = blockIdx.x * blockDim.x + threadIdx.x;
    float v = 0.0f;
    if (i < B_ * M_) v = fast_sqrtf(__uint_as_float(colmin[i]));
#pragma unroll
    for (int off = 16; off >= 1; off >>= 1)
        v += __shfl_down(v, off, 32);
    if ((threadIdx.x & 31) == 0)
        atomicAdd(out, v * (1.0f / ((float)B_ * (float)M_)));
}

// ---------------------------------------------------------------------------
extern "C" void kernel_launch(void* const* d_in, const int* in_sizes, int n_in,
                              void* d_out, int out_size, void* d_ws, size_t ws_size,
                              hipStream_t stream) {
    const float* xyz1 = (const float*)d_in[0];   // (B, N, 3) fp32
    const float* xyz2 = (const float*)d_in[1];   // (B, M, 3) fp32
    float*    out     = (float*)d_out;           // scalar fp32
    unsigned* colmin  = (unsigned*)d_ws;         // B*M uints (fp32 bit patterns)

    (void)in_sizes; (void)n_in; (void)out_size; (void)ws_size;

    const int initBlocks = (B_ * M_ + 255) / 256;
    chamfer_init<<<initBlocks, 256, 0, stream>>>(colmin, out);

    chamfer_main<<<B_ * WGS_PER_BATCH, 256, 0, stream>>>(xyz1, xyz2, colmin, out);

    chamfer_cols<<<initBlocks, 256, 0, stream>>>(colmin, out);
}